// PerturbationEncoder_56281251446890
// MI455X (gfx1250) — compile-verified
//
#include <hip/hip_runtime.h>

#define NN 100000      // N_NODES
#define NE 20000       // N_EDGES
#define D  128         // D_FEAT == HIDDEN

typedef __attribute__((ext_vector_type(2))) float v2f;
typedef __attribute__((ext_vector_type(8))) float v8f;

// ---------------- zero fill ----------------
__global__ void k_zero(float* __restrict__ p, int n) {
    int i = blockIdx.x * blockDim.x + threadIdx.x;
    if (i < n) p[i] = 0.0f;
}

// ---------------- degree counts ----------------
__global__ void k_degrees(const int* __restrict__ node_idx,
                          const int* __restrict__ edge_idx,
                          float* __restrict__ cnt_v,   // [NN]
                          float* __restrict__ cnt_e,   // [NE]
                          int nnz) {
    int stride = gridDim.x * blockDim.x;
    for (int i = blockIdx.x * blockDim.x + threadIdx.x; i < nnz; i += stride) {
        atomicAdd(&cnt_v[node_idx[i]], 1.0f);
        atomicAdd(&cnt_e[edge_idx[i]], 1.0f);
    }
}

// ------- scatter masked node features + clamped node degree into edges -------
// one wave (32 lanes) per incidence; each lane moves 4 floats (float4 gather)
__global__ __launch_bounds__(256) void k_scatter_edges(
    const float* __restrict__ feat,          // [NN, D]
    const unsigned char* __restrict__ mask,  // [NN] bool
    const int* __restrict__ node_idx,
    const int* __restrict__ edge_idx,
    const float* __restrict__ cnt_v,         // [NN]
    float* __restrict__ hsum,                // [NE, D]
    float* __restrict__ desum,               // [NE]
    int nnz) {
    int inc  = (blockIdx.x * blockDim.x + threadIdx.x) >> 5;
    int lane = threadIdx.x & 31;
    if (inc >= nnz) return;
    int v = node_idx[inc];
    int e = edge_idx[inc];
    float s = mask[v] ? 0.0f : 1.0f;
    float4 val = ((const float4*)(feat + (size_t)v * D))[lane];
    float* he = hsum + (size_t)e * D + lane * 4;
    atomicAdd(he + 0, val.x * s);
    atomicAdd(he + 1, val.y * s);
    atomicAdd(he + 2, val.z * s);
    atomicAdd(he + 3, val.w * s);
    if (lane == 0) atomicAdd(&desum[e], fmaxf(cnt_v[v], 1.0f));
}

// ------- m = (hsum @ W) * rsqrt(e_size * desum)   via V_WMMA_F32_16X16X4_F32 -------
// block = 256 threads = 8 waves; one 16-edge tile per block, one 16-col tile per wave.
__global__ __launch_bounds__(256) void k_gemm_edges(
    const float* __restrict__ hsum,   // [NE, D]
    const float* __restrict__ Wg,     // [D, D] row-major (k, n)
    const float* __restrict__ cnt_e,  // [NE]
    const float* __restrict__ desum,  // [NE]
    float* __restrict__ m) {          // [NE, D]
    __shared__ float Wlds[D * D];     // 64 KB of the 320 KB WGP LDS
    for (int i = threadIdx.x; i < (D * D) / 4; i += 256)
        ((float4*)Wlds)[i] = ((const float4*)Wg)[i];
    __syncthreads();

    const int wave  = threadIdx.x >> 5;   // 0..7 -> N tile
    const int lane  = threadIdx.x & 31;
    const int half  = lane >> 4;          // 0 | 1
    const int l16   = lane & 15;
    const int eBase = blockIdx.x * 16;
    const int nCol  = wave * 16 + l16;

    const float* arow = hsum + (size_t)(eBase + l16) * D;

    v8f acc = {};
#pragma unroll
    for (int k0 = 0; k0 < D; k0 += 4) {
        int k = k0 + half * 2;           // A: 16x4 f32, lanes 0-15 K={k0,k0+1}, 16-31 K={k0+2,k0+3}
        v2f a, bf;
        a.x  = arow[k];
        a.y  = arow[k + 1];
        bf.x = Wlds[k * D + nCol];       // B: 4x16 f32, same K split across lane halves
        bf.y = Wlds[(k + 1) * D + nCol];
        acc = __builtin_amdgcn_wmma_f32_16x16x4_f32(
            /*neg_a=*/false, a, /*neg_b=*/false, bf,
            /*c_mod=*/(short)0, acc, /*reuse_a=*/false, /*reuse_b=*/false);
    }

    // D layout: VGPR j -> row M = j + 8*half, col N = l16
#pragma unroll
    for (int j = 0; j < 8; ++j) {
        int row = eBase + j + half * 8;
        float es = fmaxf(cnt_e[row], 1.0f);
        float ds = fmaxf(desum[row], 1.0f);
        // (1/es) * rsqrt(desum/es) == rsqrt(es * desum)
        m[(size_t)row * D + nCol] = acc[j] * rsqrtf(es * ds);
    }
}

// ------- scatter edge messages back to nodes (into d_out accumulator) -------
__global__ __launch_bounds__(256) void k_scatter_nodes(
    const float* __restrict__ m,        // [NE, D]
    const int* __restrict__ node_idx,
    const int* __restrict__ edge_idx,
    float* __restrict__ out,            // [NN, D] accumulator
    int nnz) {
    int inc  = (blockIdx.x * blockDim.x + threadIdx.x) >> 5;
    int lane = threadIdx.x & 31;
    if (inc >= nnz) return;
    int v = node_idx[inc];
    int e = edge_idx[inc];
    float4 val = ((const float4*)(m + (size_t)e * D))[lane];
    float* o = out + (size_t)v * D + lane * 4;
    atomicAdd(o + 0, val.x);
    atomicAdd(o + 1, val.y);
    atomicAdd(o + 2, val.z);
    atomicAdd(o + 3, val.w);
}

// ------- out = rsqrt(d_v) * agg + b -------
__global__ void k_finalize(float* __restrict__ out,
                           const float* __restrict__ cnt_v,
                           const float* __restrict__ b,
                           int total) {
    int i = blockIdx.x * blockDim.x + threadIdx.x;
    if (i >= total) return;
    int n = i >> 7;        // / D
    int c = i & (D - 1);
    out[i] = rsqrtf(fmaxf(cnt_v[n], 1.0f)) * out[i] + b[c];
}

extern "C" void kernel_launch(void* const* d_in, const int* in_sizes, int n_in,
                              void* d_out, int out_size, void* d_ws, size_t ws_size,
                              hipStream_t stream) {
    const float*         feat     = (const float*)d_in[0];          // [NN, D]
    const float*         Wg       = (const float*)d_in[1];          // [D, D]
    const float*         bias     = (const float*)d_in[2];          // [D]
    const int*           node_idx = (const int*)d_in[3];            // [NNZ]
    const int*           edge_idx = (const int*)d_in[4];            // [NNZ]
    const unsigned char* mask     = (const unsigned char*)d_in[5];  // [NN] bool
    float* out = (float*)d_out;
    float* ws  = (float*)d_ws;
    const int nnz = in_sizes[3];

    // workspace layout (floats)
    float* cnt_v = ws;                      // NN
    float* cnt_e = ws + NN;                 // NE
    float* desum = ws + NN + NE;            // NE
    float* hsum  = ws + NN + 2 * NE;        // NE*D  (16B aligned: (NN+2*NE)*4 = 560000)
    float* mbuf  = hsum + (size_t)NE * D;   // NE*D

    // 1) zero accumulators (cnt_v|cnt_e|desum|hsum contiguous) and the output accumulator
    const int zcnt = NN + 2 * NE + NE * D;
    k_zero<<<(zcnt + 255) / 256, 256, 0, stream>>>(ws, zcnt);
    k_zero<<<(NN * D + 255) / 256, 256, 0, stream>>>(out, NN * D);

    // 2) degree counts
    k_degrees<<<2048, 256, 0, stream>>>(node_idx, edge_idx, cnt_v, cnt_e, nnz);

    // 3) scatter masked features + node degrees into edges (wave per incidence)
    {
        int blocks = (nnz + 7) / 8;  // 8 waves/block
        k_scatter_edges<<<blocks, 256, 0, stream>>>(feat, mask, node_idx, edge_idx,
                                                    cnt_v, hsum, desum, nnz);
    }

    // 4) f32 WMMA GEMM with fused UniGCN normalizers
    k_gemm_edges<<<NE / 16, 256, 0, stream>>>(hsum, Wg, cnt_e, desum, mbuf);

    // 5) scatter edge messages back to nodes
    {
        int blocks = (nnz + 7) / 8;
        k_scatter_nodes<<<blocks, 256, 0, stream>>>(mbuf, node_idx, edge_idx, out, nnz);
    }

    // 6) final scale + bias
    k_finalize<<<(NN * D + 255) / 256, 256, 0, stream>>>(out, cnt_v, bias, NN * D);
}